// Attention_50989851738305
// MI455X (gfx1250) — compile-verified
//
#include <hip/hip_runtime.h>
#include <hip/hip_bf16.h>

typedef __attribute__((ext_vector_type(2))) float v2f;
typedef __attribute__((ext_vector_type(4))) float v4f;
typedef __attribute__((ext_vector_type(8))) float v8f;

#define BATCH   2
#define SEQ     2048
#define DIM     1024
#define CHUNKS  64
#define ROWS_PER_CHUNK (SEQ / CHUNKS)   // 32

// ---------------------------------------------------------------------------
// Kernel 1: partial sums of hidden over S-chunks.
// partials[b][c][d] = sum_{r in chunk c} hidden[b][r][d]
// grid (DIM/256, CHUNKS, BATCH), block 256. Fully coalesced along d.
// ---------------------------------------------------------------------------
__global__ __launch_bounds__(256) void partial_sum_kernel(
    const float* __restrict__ hs, float* __restrict__ partials) {
  const int d = blockIdx.x * 256 + threadIdx.x;
  const int c = blockIdx.y;
  const int b = blockIdx.z;
  const float* p = hs + ((size_t)b * SEQ + (size_t)c * ROWS_PER_CHUNK) * DIM + d;
  float s = 0.f;
#pragma unroll 8
  for (int r = 0; r < ROWS_PER_CHUNK; ++r) s += p[(size_t)r * DIM];
  partials[((size_t)b * CHUNKS + c) * DIM + d] = s;
}

// ---------------------------------------------------------------------------
// Kernel 2: finalize mean and zero-pad to a 16 x 1024 A-matrix.
// h_pad[row][d] = (row < 2) ? mean_s hidden[row][s][d] : 0
// ---------------------------------------------------------------------------
__global__ __launch_bounds__(256) void finalize_mean_kernel(
    const float* __restrict__ partials, float* __restrict__ h_pad) {
  const int i = blockIdx.x * 256 + threadIdx.x;  // 0 .. 16*1024-1
  const int row = i >> 10;
  const int d = i & (DIM - 1);
  float v = 0.f;
  if (row < BATCH) {
    const float* p = partials + ((size_t)row * CHUNKS) * DIM + d;
    float s = 0.f;
#pragma unroll
    for (int c = 0; c < CHUNKS; ++c) s += p[(size_t)c * DIM];
    v = s * (1.0f / (float)SEQ);
  }
  h_pad[i] = v;
}

// ---------------------------------------------------------------------------
// Kernel 3/4: D[16][1024] = A[16][1024] @ W^T  via V_WMMA_F32_16X16X4_F32.
//   D[m][n] = sum_k A[m][k] * W[n][k]   (B-matrix[k][n] = W[n][k])
// One wave (32 threads) per 16-wide N tile; K loop in steps of 4.
// VGPR layouts per CDNA5 ISA 7.12.2:
//   A 16x4 : lane m=lane%16; vgpr j holds K = k0 + 2*half + j   (half=lane/16)
//   B 4x16 : lane n=lane%16; vgpr j holds K = k0 + 2*half + j   (rows j, j+2)
//   D 16x16: vgpr j -> row (j + 8*half), col = n0 + lane%16
// Per-lane loads for both A and B are contiguous float2 -> global_load_b64.
// ---------------------------------------------------------------------------
__global__ __launch_bounds__(32) void gemv16_wmma_kernel(
    const float* __restrict__ A,   // [16][DIM] row-major, rows >= BATCH are zero
    const float* __restrict__ W,   // [DIM][DIM] row-major
    float* __restrict__ Dout) {    // [16][DIM]
  const int lane = threadIdx.x;
  const int m    = lane & 15;
  const int half = lane >> 4;          // 0 or 1
  const int n0   = blockIdx.x * 16;

  const float* aPtr = A + (size_t)m * DIM + 2 * half;
  const float* bPtr = W + (size_t)(n0 + m) * DIM + 2 * half;

  v8f c = {};
#pragma unroll 4
  for (int k0 = 0; k0 < DIM; k0 += 4) {
    v2f a = *(const v2f*)(aPtr + k0);
    v2f b = *(const v2f*)(bPtr + k0);
    // 8 args: (neg_a, A, neg_b, B, c_mod, C, reuse_a, reuse_b)
    c = __builtin_amdgcn_wmma_f32_16x16x4_f32(
        false, a, false, b, (short)0, c, false, false);
  }

#pragma unroll
  for (int j = 0; j < 8; ++j) {
    Dout[(size_t)(j + half * 8) * DIM + n0 + m] = c[j];
  }
}

// ---------------------------------------------------------------------------
// Kernel 5: broadcast the two per-batch output rows across all S positions.
// out[b][s][:] = o_pad[b][:]  — float4 stores, fully coalesced, ~16 MB.
// ---------------------------------------------------------------------------
__global__ __launch_bounds__(256) void broadcast_kernel(
    const float* __restrict__ o_pad, float* __restrict__ out) {
  const size_t i = (size_t)blockIdx.x * 256 + threadIdx.x;  // one float4 each
  const int d4 = (int)(i & (DIM / 4 - 1));                  // 0..255
  const size_t bs = i >> 8;                                 // b*SEQ + s
  const int b = (int)(bs >> 11);                            // SEQ = 2048
  v4f val = *(const v4f*)(o_pad + (size_t)b * DIM + (size_t)d4 * 4);
  *(v4f*)(out + i * 4) = val;
}

// ---------------------------------------------------------------------------
extern "C" void kernel_launch(void* const* d_in, const int* in_sizes, int n_in,
                              void* d_out, int out_size, void* d_ws, size_t ws_size,
                              hipStream_t stream) {
  (void)in_sizes; (void)n_in; (void)out_size; (void)ws_size;
  const float* hs = (const float*)d_in[0];
  const float* Wq = (const float*)d_in[1];  (void)Wq;  // dead: scores are fully masked
  const float* Wk = (const float*)d_in[2];  (void)Wk;  // dead: scores are fully masked
  const float* Wv = (const float*)d_in[3];
  const float* Wo = (const float*)d_in[4];
  float* out = (float*)d_out;

  char* ws = (char*)d_ws;
  float* partials = (float*)(ws);                        // 2*64*1024 f32 = 512 KB
  float* h_pad    = (float*)(ws + (512 << 10));          // 16*1024 f32  =  64 KB
  float* v_pad    = (float*)(ws + (576 << 10));          // 16*1024 f32  =  64 KB
  float* o_pad    = (float*)(ws + (640 << 10));          // 16*1024 f32  =  64 KB

  // 1) mean over S (two-phase deterministic reduction, ~16 MB read)
  partial_sum_kernel<<<dim3(DIM / 256, CHUNKS, BATCH), 256, 0, stream>>>(hs, partials);
  finalize_mean_kernel<<<(16 * DIM) / 256, 256, 0, stream>>>(partials, h_pad);

  // 2) v_mean = h_mean @ Wv^T   (16x1024x1024 f32 WMMA GEMM)
  gemv16_wmma_kernel<<<DIM / 16, 32, 0, stream>>>(h_pad, Wv, v_pad);

  // 3) out_row = v_mean @ Wo^T  (16x1024x1024 f32 WMMA GEMM)
  gemv16_wmma_kernel<<<DIM / 16, 32, 0, stream>>>(v_pad, Wo, o_pad);

  // 4) broadcast rows across the sequence (~16 MB write)
  broadcast_kernel<<<(BATCH * SEQ * DIM / 4) / 256, 256, 0, stream>>>(o_pad, out);
}